// TimeAffineFlow_84327387889964
// MI455X (gfx1250) — compile-verified
//
#include <hip/hip_runtime.h>
#include <hip/hip_bf16.h>

typedef __attribute__((ext_vector_type(2))) float v2f;
typedef __attribute__((ext_vector_type(8))) float v8f;

#define NTERMS 12   // residual ~ ||tM||^13/13! ~ 5e-10, below fp32 eps

// One wave (32 lanes) owns a 16-row batch tile held in a private LDS buffer.
// Each Taylor step: W(16x64) = V(16x64) * Mt(64x64) via v_wmma_f32_16x16x4_f32,
// then per-row scale by t_m/k, accumulate into Y, write term back to LDS.
__global__ __launch_bounds__(256) void taf_expm_taylor_kernel(
    const float* __restrict__ t,   // (B,)
    const float* __restrict__ x,   // (B, 63)
    const float* __restrict__ A,   // (62, 62)
    const float* __restrict__ b,   // (62,)
    const float* __restrict__ c,   // (62,)
    float* __restrict__ out)       // (B, 63)
{
    __shared__ float Mt[64 * 64];       // Mt[k*64+n] = M[n][k]
    __shared__ float V[8][16 * 64];     // per-wave current Taylor term (row-major 16x64)

    const int tid  = threadIdx.x;
    const int wave = tid >> 5;
    const int lane = tid & 31;
    const int g    = lane >> 4;         // half-wave group (0/1)
    const int ln   = lane & 15;
    const int tileBase = blockIdx.x * 128 + wave * 16;  // first batch row of this tile

    // Reciprocal table: keeps 1/k out of the inner loop (no v_rcp/v_div chain).
    const float invk_tab[NTERMS + 1] = {
        0.0f, 1.0f, 1.0f / 2, 1.0f / 3, 1.0f / 4, 1.0f / 5, 1.0f / 6,
        1.0f / 7, 1.0f / 8, 1.0f / 9, 1.0f / 10, 1.0f / 11, 1.0f / 12
    };

    // ---- Build Mt = M^T cooperatively -------------------------------------
    // M: row 0 = e_63 ; rows 1..62 = [c | A | b] ; row 63 = 0
    for (int idx = tid; idx < 64 * 64; idx += 256) {
        const int k = idx >> 6;         // column of M
        const int n = idx & 63;         // row of M
        float val = 0.0f;
        if (n >= 1 && n <= 62) {
            const int i = n - 1;
            if (k == 0)        val = c[i];
            else if (k <= 62)  val = A[i * 62 + (k - 1)];
            else               val = b[i];
        } else if (n == 0 && k == 63) {
            val = 1.0f;
        }
        Mt[idx] = val;
    }

    // ---- Load x tile (+ homogeneous 1.0 in col 63) into this wave's V -----
    float* Vw = &V[wave][0];
    for (int idx = lane; idx < 16 * 64; idx += 32) {
        const int m = idx >> 6;
        const int n = idx & 63;
        Vw[idx] = (n < 63) ? x[(size_t)(tileBase + m) * 63 + n] : 1.0f;
    }

    __syncthreads();

    // ---- Per-lane t values: this lane scales rows m = r + 8*g (C/D layout) --
    float tv[8];
#pragma unroll
    for (int r = 0; r < 8; ++r) tv[r] = t[tileBase + r + 8 * g];

    // ---- Preload all B-fragments of Mt into registers (constant all steps) --
    // B frag (nb,kb): reg j holds Mt[k][n], k = 4*kb + 2*g + j, n = nb*16 + ln
    v2f Bf[4][16];
#pragma unroll
    for (int nb = 0; nb < 4; ++nb) {
#pragma unroll
        for (int kb = 0; kb < 16; ++kb) {
            const int kb0 = 4 * kb + 2 * g;
            const int n   = nb * 16 + ln;
            v2f f;
            f[0] = Mt[kb0 * 64 + n];
            f[1] = Mt[(kb0 + 1) * 64 + n];
            Bf[nb][kb] = f;
        }
    }

    // ---- Y accumulator = term 0 (xpad), read in C/D layout ------------------
    v8f Y[4];
#pragma unroll
    for (int nb = 0; nb < 4; ++nb) {
#pragma unroll
        for (int r = 0; r < 8; ++r)
            Y[nb][r] = Vw[(r + 8 * g) * 64 + nb * 16 + ln];
    }

    // ---- Taylor iterations --------------------------------------------------
    for (int k = 1; k <= NTERMS; ++k) {
        v8f acc[4];
        acc[0] = (v8f){}; acc[1] = (v8f){}; acc[2] = (v8f){}; acc[3] = (v8f){};

#pragma unroll
        for (int kb = 0; kb < 16; ++kb) {
            // A fragment: reg j holds V[m][kk+j], m = ln, kk = 4*kb + 2*g (even -> b64)
            const int kk = 4 * kb + 2 * g;
            v2f Af;
            Af[0] = Vw[ln * 64 + kk];
            Af[1] = Vw[ln * 64 + kk + 1];
            acc[0] = __builtin_amdgcn_wmma_f32_16x16x4_f32(false, Af, false, Bf[0][kb],
                                                           (short)0, acc[0], false, false);
            acc[1] = __builtin_amdgcn_wmma_f32_16x16x4_f32(false, Af, false, Bf[1][kb],
                                                           (short)0, acc[1], false, false);
            acc[2] = __builtin_amdgcn_wmma_f32_16x16x4_f32(false, Af, false, Bf[2][kb],
                                                           (short)0, acc[2], false, false);
            acc[3] = __builtin_amdgcn_wmma_f32_16x16x4_f32(false, Af, false, Bf[3][kb],
                                                           (short)0, acc[3], false, false);
        }

        // term_k = (t_m / k) * W ; Y += term_k ; write term_k back for next step
        const float invk = invk_tab[k];
#pragma unroll
        for (int nb = 0; nb < 4; ++nb) {
#pragma unroll
            for (int r = 0; r < 8; ++r) {
                const float term = acc[nb][r] * (tv[r] * invk);
                Y[nb][r] += term;
                Vw[(r + 8 * g) * 64 + nb * 16 + ln] = term;   // same-wave DS ordering
            }
        }
    }

    // ---- Store y[:, 0:63] (drop homogeneous column 63) ----------------------
#pragma unroll
    for (int nb = 0; nb < 4; ++nb) {
        const int n = nb * 16 + ln;
#pragma unroll
        for (int r = 0; r < 8; ++r) {
            if (n < 63)
                out[(size_t)(tileBase + r + 8 * g) * 63 + n] = Y[nb][r];
        }
    }
}

extern "C" void kernel_launch(void* const* d_in, const int* in_sizes, int n_in,
                              void* d_out, int out_size, void* d_ws, size_t ws_size,
                              hipStream_t stream) {
    const float* t = (const float*)d_in[0];   // (16384,)
    const float* x = (const float*)d_in[1];   // (16384, 63)
    const float* A = (const float*)d_in[2];   // (62, 62)
    const float* b = (const float*)d_in[3];   // (62,)
    const float* c = (const float*)d_in[4];   // (62,)
    float* out = (float*)d_out;               // (16384, 63)

    const int B = in_sizes[0];                // 16384
    const int blocks = B / 128;               // 128 rows per block (8 waves x 16)

    taf_expm_taylor_kernel<<<blocks, 256, 0, stream>>>(t, x, A, b, c, out);
}